// sage_layer_50706383897204
// MI455X (gfx1250) — compile-verified
//
#include <hip/hip_runtime.h>
#include <hip/hip_bf16.h>

typedef __attribute__((ext_vector_type(2))) float v2f;
typedef __attribute__((ext_vector_type(8))) float v8f;

#define D 128          // D_IN == D_OUT == 128
#define TWO_D 256

// ---------------------------------------------------------------------------
// Phase 0: zero the support accumulator in workspace (N*128 floats).
// ---------------------------------------------------------------------------
__global__ void gcn_zero_kernel(float4* __restrict__ p, long n4) {
    long i = (long)blockIdx.x * blockDim.x + threadIdx.x;
    if (i < n4) {
        p[i] = make_float4(0.f, 0.f, 0.f, 0.f);
    }
}

// ---------------------------------------------------------------------------
// Phase 1: edge scatter. One wave (32 lanes) per edge; each lane handles 4
// contiguous columns -> float4 gather from x[src], scaled by edge_val, then
// 4 fp32 global atomic adds into support[dst]. x and support both fit in the
// 192MB L2, so gathers and L2-side atomics stay on-chip.
// ---------------------------------------------------------------------------
__global__ void gcn_scatter_kernel(const float* __restrict__ x,
                                   const int*   __restrict__ edge_src,
                                   const int*   __restrict__ edge_dst,
                                   const float* __restrict__ edge_val,
                                   float*       __restrict__ support,
                                   int nEdges) {
    long gtid = (long)blockIdx.x * blockDim.x + threadIdx.x;
    int e    = (int)(gtid >> 5);
    int lane = (int)(gtid & 31);
    if (e >= nEdges) return;

    int   src = edge_src[e];          // uniform across the wave -> broadcast
    int   dst = edge_dst[e];
    float val = edge_val[e];

    const float4* xr = (const float4*)(x + (long)src * D);
    float4 g = xr[lane];              // cols [4*lane, 4*lane+3]

    float* s = support + (long)dst * D + lane * 4;
    unsafeAtomicAdd(s + 0, g.x * val);    // global_atomic_add_f32 (no return)
    unsafeAtomicAdd(s + 1, g.y * val);
    unsafeAtomicAdd(s + 2, g.z * val);
    unsafeAtomicAdd(s + 3, g.w * val);
}

// ---------------------------------------------------------------------------
// Phase 2: out[N,128] = x @ W[0:128,:] + support @ W[128:256,:] + b
// using V_WMMA_F32_16X16X4_F32 (exact fp32 path on CDNA5).
// One wave -> one 16x16 output tile. Block = 8 waves covering all 8 column
// tiles of one 16-row panel (A panel shared via L0). N % 16 == 0 so EXEC is
// all-ones everywhere (WMMA requirement).
//
// Fragment layouts (ISA 7.12.2, 32-bit data):
//   A 16x4 : lane<16 -> (M=lane, K=0..1), lane>=16 -> (M=lane-16, K=2..3)
//   B 4x16 : VGPR0 = {K0 | K2}, VGPR1 = {K1 | K3} across half-waves
//   C 16x16: VGPR v -> row v (lanes 0-15) / v+8 (lanes 16-31), col = lane&15
// ---------------------------------------------------------------------------
__global__ void gcn_wmma_gemm_kernel(const float* __restrict__ x,
                                     const float* __restrict__ support,
                                     const float* __restrict__ W,   // [256,128]
                                     const float* __restrict__ bias,
                                     float*       __restrict__ out,
                                     int N) {
    const int lane  = threadIdx.x & 31;
    const int wave  = threadIdx.x >> 5;       // 0..7 -> column tile
    const int row0  = blockIdx.x * 16;
    const int col0  = wave * 16;
    const int l15   = lane & 15;
    const int khalf = (lane >> 4) << 1;       // 0 for lanes 0-15, 2 for 16-31

    v8f c = {};

    const float* arow0 = x       + (long)(row0 + l15) * D + khalf;
    const float* arow1 = support + (long)(row0 + l15) * D + khalf;
    const float* wcol  = W + col0 + l15;      // column (col0+l15) of W

    // out += x @ W[0:128, :]
    #pragma unroll 8
    for (int k = 0; k < D; k += 4) {
        v2f a = *(const v2f*)(arow0 + k);     // 8B-aligned float2
        v2f b;
        b.x = wcol[(long)(k + khalf)     * D];
        b.y = wcol[(long)(k + khalf + 1) * D];
        c = __builtin_amdgcn_wmma_f32_16x16x4_f32(
                /*neg_a=*/false, a, /*neg_b=*/false, b,
                /*c_mod=*/(short)0, c, /*reuse_a=*/false, /*reuse_b=*/false);
    }
    // out += support @ W[128:256, :]
    #pragma unroll 8
    for (int k = 0; k < D; k += 4) {
        v2f a = *(const v2f*)(arow1 + k);
        v2f b;
        b.x = wcol[(long)(D + k + khalf)     * D];
        b.y = wcol[(long)(D + k + khalf + 1) * D];
        c = __builtin_amdgcn_wmma_f32_16x16x4_f32(
                false, a, false, b, (short)0, c, false, false);
    }

    const float bv      = bias[col0 + l15];
    const int   rowbase = row0 + ((lane >> 4) << 3);   // +8 for upper half-wave
    #pragma unroll
    for (int v = 0; v < 8; ++v) {
        out[(long)(rowbase + v) * D + col0 + l15] = c[v] + bv;
    }
}

// ---------------------------------------------------------------------------
// Launch: zero -> scatter -> wmma-gemm, all serialized on `stream`.
// Inputs: x[N*128] f32, edge_src[E] i32, edge_dst[E] i32, edge_val[E] f32,
//         W[256*128] f32, b[128] f32.  Output: out[N*128] f32.
// Workspace: support = N*128 floats (51.2 MB) at the head of d_ws.
// ---------------------------------------------------------------------------
extern "C" void kernel_launch(void* const* d_in, const int* in_sizes, int n_in,
                              void* d_out, int out_size, void* d_ws, size_t ws_size,
                              hipStream_t stream) {
    const float* x        = (const float*)d_in[0];
    const int*   edge_src = (const int*)  d_in[1];
    const int*   edge_dst = (const int*)  d_in[2];
    const float* edge_val = (const float*)d_in[3];
    const float* W        = (const float*)d_in[4];
    const float* bias     = (const float*)d_in[5];
    float*       out      = (float*)d_out;

    const int N = in_sizes[0] / D;    // 100000 (divisible by 16)
    const int E = in_sizes[1];        // 1600000

    float* support = (float*)d_ws;    // N*128 floats

    // Phase 0: zero support
    long n4 = (long)N * (D / 4);
    int zb = (int)((n4 + 255) / 256);
    gcn_zero_kernel<<<zb, 256, 0, stream>>>((float4*)support, n4);

    // Phase 1: scatter-add (wave per edge)
    long st = (long)E * 32;
    int sb = (int)((st + 255) / 256);
    gcn_scatter_kernel<<<sb, 256, 0, stream>>>(x, edge_src, edge_dst, edge_val,
                                               support, E);

    // Phase 2: fused concat+GEMM via fp32 WMMA
    gcn_wmma_gemm_kernel<<<N / 16, 256, 0, stream>>>(x, support, W, bias, out, N);
}